// ModelNew_67276367724709
// MI455X (gfx1250) — compile-verified
//
#include <hip/hip_runtime.h>
#include <hip/hip_bf16.h>
#include <math.h>

// Problem constants (from reference): x[M=16384, N=4096], weight[O=16384, N=4096]
#define NN 4096
#define OO 16384
#define CHUNKS 64                 // row chunks for the weight column-sum
#define ROWS_PER_CHUNK (OO / CHUNKS)   // 256

typedef __attribute__((ext_vector_type(2))) float v2f;
typedef __attribute__((ext_vector_type(8))) float v8f;

// ---------------------------------------------------------------------------
// Stage 1: partial column sums of weight via WMMA (ones^T @ W_chunk).
// A (16x4) = all ones -> D[i][j] = sum_k B[k][j]  (column sums of the 4x16 tile).
// B layout (f32 4x16, wave32): lane l holds column n = l&15; lanes 0-15 carry
// K=0 (VGPR0) / K=1 (VGPR1); lanes 16-31 carry K=2 / K=3.
// Each wave owns 16 columns and marches down 256 rows (64 WMMAs).
// Deterministic: each (chunk, column) partial is written exactly once.
// ---------------------------------------------------------------------------
__global__ void k_colsum_wmma(const float* __restrict__ W,
                              float* __restrict__ partial) {
    const int lane = threadIdx.x & 31;
    const int wave = threadIdx.x >> 5;
    const int colTile = blockIdx.x * 8 + wave;      // 8 waves/block, 32 blocks.x -> 256 tiles
    const int col0 = colTile * 16;
    const int rowBase = blockIdx.y * ROWS_PER_CHUNK;

    const int n  = lane & 15;
    const int kh = (lane >> 4) << 1;                // 0 or 2

    v8f acc = {};
    v2f a;
    a.x = 1.0f; a.y = 1.0f;                         // ones: layout-invariant

    const float* p = W + (size_t)(rowBase + kh) * NN + col0 + n;
#pragma unroll 4
    for (int r = 0; r < ROWS_PER_CHUNK; r += 4) {
        v2f b;
        b.x = p[0];                                 // row rowBase+r+kh
        b.y = p[NN];                                // row rowBase+r+kh+1
        acc = __builtin_amdgcn_wmma_f32_16x16x4_f32(
            false, a, false, b, (short)0, acc, false, false);
        p += (size_t)4 * NN;
    }

    // acc[0] at lane l = column sum of column (l&15); lanes 0-15 suffice.
    if (lane < 16) {
        partial[(size_t)blockIdx.y * NN + col0 + lane] = acc[0];
    }
}

// ---------------------------------------------------------------------------
// Stage 2a: fixed-order reduction of partials -> c[N]. 16 blocks x 256 threads.
// ---------------------------------------------------------------------------
__global__ void k_reduce_c(const float* __restrict__ partial,
                           float* __restrict__ c) {
    const int j = blockIdx.x * 256 + threadIdx.x;
    float s = 0.0f;
#pragma unroll 8
    for (int k = 0; k < CHUNKS; ++k) s += partial[(size_t)k * NN + j];
    c[j] = s;
}

// ---------------------------------------------------------------------------
// Stage 2b: col_sum_bias = sum(bias - subtract). One block, deterministic.
// ---------------------------------------------------------------------------
__global__ void k_bias_sum(const float* __restrict__ bias,
                           const float* __restrict__ sub,
                           float* __restrict__ out_s) {
    __shared__ float sm[256];
    float acc = 0.0f;
    for (int i = threadIdx.x; i < OO; i += 256) acc += bias[i] - sub[i];
    sm[threadIdx.x] = acc;
    __syncthreads();
    for (int s = 128; s > 0; s >>= 1) {
        if (threadIdx.x < s) sm[threadIdx.x] += sm[threadIdx.x + s];
        __syncthreads();
    }
    if (threadIdx.x == 0) *out_s = sm[0];
}

// ---------------------------------------------------------------------------
// Stage 3: one block per row. Row kept in registers (4 x float4 per thread),
// dot against L2-resident c, LDS tree reduce, GELU(tanh) on the row mean,
// broadcast, residual add, single write. x is read exactly once.
// ---------------------------------------------------------------------------
__global__ void k_row_fused(const float* __restrict__ x,
                            const float* __restrict__ c,
                            const float* __restrict__ s_ptr,
                            float* __restrict__ out) {
    __shared__ float sm[256];
    __shared__ float gsh;
    const int t = threadIdx.x;
    const size_t row = blockIdx.x;

    const float4* x4 = (const float4*)(x + row * (size_t)NN);
    const float4* c4 = (const float4*)c;

    float4 xs[4];
    float dot = 0.0f;
#pragma unroll
    for (int q = 0; q < 4; ++q) {
        const float4 xv = x4[q * 256 + t];
        const float4 cv = c4[q * 256 + t];
        xs[q] = xv;
        dot += xv.x * cv.x + xv.y * cv.y + xv.z * cv.z + xv.w * cv.w;
    }

    sm[t] = dot;
    __syncthreads();
    for (int s = 128; s > 0; s >>= 1) {
        if (t < s) sm[t] += sm[t + s];
        __syncthreads();
    }
    if (t == 0) {
        const float m = (sm[0] + *s_ptr) * (1.0f / (float)OO);
        const float inner = 0.7978845608028654f * (m + 0.044715f * m * m * m);
        gsh = 0.5f * m * (1.0f + tanhf(inner));
    }
    __syncthreads();
    const float g = gsh;

    float4* o4 = (float4*)(out + row * (size_t)NN);
#pragma unroll
    for (int q = 0; q < 4; ++q) {
        float4 v = xs[q];
        v.x += g; v.y += g; v.z += g; v.w += g;
        o4[q * 256 + t] = v;
    }
}

extern "C" void kernel_launch(void* const* d_in, const int* in_sizes, int n_in,
                              void* d_out, int out_size, void* d_ws, size_t ws_size,
                              hipStream_t stream) {
    const float* x      = (const float*)d_in[0];   // [M, 4096]
    const float* weight = (const float*)d_in[1];   // [16384, 4096]
    const float* bias   = (const float*)d_in[2];   // [16384]
    const float* sub    = (const float*)d_in[3];   // [16384]
    // d_in[4] = out_features scalar (16384), compile-time constant here.

    const int M = in_sizes[0] / NN;                // 16384

    // Workspace layout (floats): partial[CHUNKS*NN] | c[NN] | s[1]  (~1.05 MB)
    float* partial = (float*)d_ws;
    float* c       = partial + (size_t)CHUNKS * NN;
    float* s       = c + NN;
    float* out     = (float*)d_out;

    // Stage 1: weight column partial sums (WMMA ones-trick), 32x64 blocks x 8 waves.
    dim3 g1(NN / 16 / 8, CHUNKS);
    k_colsum_wmma<<<g1, 256, 0, stream>>>(weight, partial);

    // Stage 2: deterministic reductions.
    k_bias_sum<<<1, 256, 0, stream>>>(bias, sub, s);
    k_reduce_c<<<NN / 256, 256, 0, stream>>>(partial, c);

    // Stage 3: fused per-row dot + GELU + residual (single pass over x).
    k_row_fused<<<M, 256, 0, stream>>>(x, c, s, out);
}